// full_attns_34402688041208
// MI455X (gfx1250) — compile-verified
//
#include <hip/hip_runtime.h>

typedef float v2f __attribute__((ext_vector_type(2)));
typedef float v4f __attribute__((ext_vector_type(4)));
typedef float v8f __attribute__((ext_vector_type(8)));

#define NSEQ 2048
#define FDIM 64
#define TDIM 12
#define HID  32

__device__ __forceinline__ float gelu_exact(float x) {
    // jax.nn.gelu(approximate=False): 0.5*x*(1+erf(x/sqrt(2)))
    return 0.5f * x * (1.0f + erff(x * 0.70710678118654752f));
}

// ---------------------------------------------------------------------------
// Kernel 1: per 16 rows (row = b*N+n):
//   q[m][f]   = gelu( sum_t x[row_m, f, t] * w1[t] )          (LDS tile 16x64)
//   U[f][n]   = sum_d W2[f, h*8+d] * a[(n<4?0:8)+d], h=n&3    (64x8, pad to 16)
//   S[row][n] = sum_f q[row][f] * U[f][n]  via 16x V_WMMA_F32_16X16X4_F32
// S[row][0..3] = sq(h), S[row][4..7] = sk(h).
// ---------------------------------------------------------------------------
__global__ __launch_bounds__(256) void k1_qproj(
    const float* __restrict__ x, const float* __restrict__ w1,
    const float* __restrict__ W2, const float* __restrict__ a,
    float* __restrict__ S)
{
    __shared__ float q_lds[16 * 68];   // row stride 68 to dodge bank conflicts
    __shared__ float U_lds[64 * 16];   // cols 8..15 zero

    const int tid = threadIdx.x;
    const int r0  = blockIdx.x * 16;   // 256 blocks * 16 rows = 4096 rows

    // w1 into registers (broadcast loads)
    float w1r[TDIM];
    #pragma unroll
    for (int t = 0; t < TDIM; ++t) w1r[t] = w1[t];

    // U: 512 values + 512 zero-pads; 2+2 per thread
    #pragma unroll
    for (int it = 0; it < 2; ++it) {
        int idx = tid * 2 + it;              // 0..511
        int f = idx >> 3;
        int n = idx & 7;
        const float* av = (n < 4) ? a : (a + 8);
        int h = n & 3;
        float s = 0.f;
        #pragma unroll
        for (int dd = 0; dd < 8; ++dd)
            s += W2[f * HID + h * 8 + dd] * av[dd];
        U_lds[f * 16 + n]     = s;
        U_lds[f * 16 + 8 + n] = 0.f;
    }

    // q tile: 16 rows x 64 f = 1024 values, 4 per thread; x row chunk is
    // 12 consecutive floats, 16B-aligned (f*48 bytes) -> 3x float4.
    #pragma unroll
    for (int it = 0; it < 4; ++it) {
        int v = tid + 256 * it;              // 0..1023
        int m = v >> 6;
        int f = v & 63;
        const float* xp = x + (size_t)(r0 + m) * (FDIM * TDIM) + f * TDIM;
        v4f x0 = *(const v4f*)(xp);
        v4f x1 = *(const v4f*)(xp + 4);
        v4f x2 = *(const v4f*)(xp + 8);
        float s = x0.x*w1r[0] + x0.y*w1r[1] + x0.z*w1r[2]  + x0.w*w1r[3]
                + x1.x*w1r[4] + x1.y*w1r[5] + x1.z*w1r[6]  + x1.w*w1r[7]
                + x2.x*w1r[8] + x2.y*w1r[9] + x2.z*w1r[10] + x2.w*w1r[11];
        q_lds[m * 68 + f] = gelu_exact(s);
    }
    __syncthreads();

    // Wave 0 performs the WMMA chain (EXEC all-ones within the wave).
    if (tid < 32) {
        const int lane = tid;
        const int m    = lane & 15;          // M index (lanes 0-15 / 16-31)
        const int half = lane >> 4;          // selects K pair per ISA layout

        v8f acc = {};
        const float* qrow = q_lds + m * 68;
        #pragma unroll
        for (int kk = 0; kk < 16; ++kk) {
            int f0 = kk * 4 + half * 2;      // A: VGPR j -> K = f0 + j
            v2f A; A.x = qrow[f0];  A.y = qrow[f0 + 1];
            v2f Bm; Bm.x = U_lds[f0 * 16 + m]; Bm.y = U_lds[(f0 + 1) * 16 + m];
            acc = __builtin_amdgcn_wmma_f32_16x16x4_f32(
                false, A, false, Bm, (short)0, acc, false, false);
        }

        // C/D layout: VGPR r, lanes 0-15 -> M=r; lanes 16-31 -> M=8+r; N=lane&15
        if (m < 8) {
            int rowb = r0 + half * 8;
            #pragma unroll
            for (int r = 0; r < 8; ++r)
                S[(size_t)(rowb + r) * 8 + m] = acc[r];
        }
    }
}

// ---------------------------------------------------------------------------
// Kernel 2: out[b,h,i,j] = gelu(sq[b,i,h] + sk[b,j,h] + adj[i,j]*aadj)
// grid = (2, 2048): blockIdx.y = i, blockIdx.x picks 1024-wide j chunk.
// Each thread: 4 consecutive j, loops over (b,h) = 8 float4 NT stores.
// ---------------------------------------------------------------------------
__global__ __launch_bounds__(256) void k2_scores(
    const float* __restrict__ adj, const float* __restrict__ S,
    const float* __restrict__ a, float* __restrict__ out)
{
    __shared__ float sk_lds[2 * 1024 * 4];   // [b][jlocal][h], 32KB

    const int tid = threadIdx.x;
    const int i  = blockIdx.y;
    const int j0 = blockIdx.x * 1024;

    // Stage sk for this j chunk (coalesced float4 loads of S cols 4..7)
    #pragma unroll
    for (int b = 0; b < 2; ++b) {
        #pragma unroll
        for (int rep = 0; rep < 4; ++rep) {
            int jl = rep * 256 + tid;
            v4f v = *(const v4f*)(S + (size_t)(b * NSEQ + j0 + jl) * 8 + 4);
            *(v4f*)(sk_lds + (b * 1024 + jl) * 4) = v;
        }
    }
    v4f sqv0 = *(const v4f*)(S + (size_t)(0 * NSEQ + i) * 8);
    v4f sqv1 = *(const v4f*)(S + (size_t)(1 * NSEQ + i) * 8);
    float aadj = a[16];
    __syncthreads();

    const int jt = j0 + tid * 4;
    v4f adj4 = *(const v4f*)(adj + (size_t)i * NSEQ + jt);
    adj4.x *= aadj; adj4.y *= aadj; adj4.z *= aadj; adj4.w *= aadj;

    #pragma unroll
    for (int b = 0; b < 2; ++b) {
        v4f sqv = (b == 0) ? sqv0 : sqv1;
        const float* skp0 = sk_lds + (b * 1024 + tid * 4) * 4;
        #pragma unroll
        for (int h = 0; h < 4; ++h) {
            float base = sqv[h];
            v4f o;
            o.x = gelu_exact(base + skp0[h]      + adj4.x);
            o.y = gelu_exact(base + skp0[4 + h]  + adj4.y);
            o.z = gelu_exact(base + skp0[8 + h]  + adj4.z);
            o.w = gelu_exact(base + skp0[12 + h] + adj4.w);
            float* op = out + ((size_t)(b * 4 + h) * NSEQ + i) * NSEQ + jt;
            __builtin_nontemporal_store(o, (v4f*)op);
        }
    }
}

extern "C" void kernel_launch(void* const* d_in, const int* in_sizes, int n_in,
                              void* d_out, int out_size, void* d_ws, size_t ws_size,
                              hipStream_t stream) {
    const float* x   = (const float*)d_in[0];   // (2,2048,64,12)
    const float* adj = (const float*)d_in[1];   // (2048,2048)
    const float* w1  = (const float*)d_in[2];   // (12,)
    const float* W2  = (const float*)d_in[3];   // (64,32)
    const float* a   = (const float*)d_in[4];   // (17,)
    float* out = (float*)d_out;                 // (2,4,2048,2048)
    float* S   = (float*)d_ws;                  // 4096*8 floats = 128KB scratch

    k1_qproj<<<dim3(256), dim3(256), 0, stream>>>(x, w1, W2, a, S);
    k2_scores<<<dim3(2, 2048), dim3(256), 0, stream>>>(adj, S, a, out);
}